// TokenToWordAggregator_8615704396314
// MI455X (gfx1250) — compile-verified
//
#include <hip/hip_runtime.h>
#include <math.h>
#include <stdint.h>

// Per-word attention pooling over ragged sorted segments, fully fused.
// One wave32 per word. H = 768 = 32 lanes * 6 chunks * 4 floats.
// CDNA5 path: weights staged into LDS via GLOBAL_LOAD_ASYNC_TO_LDS_B128
// (ASYNCcnt + s_wait_asynccnt), features streamed with coalesced b128 loads,
// output written with non-temporal b128 stores.

#define HDIM    768
#define CHUNKS  6          // 6 * 128 = 768
#define CSTRIDE 128        // 32 lanes * float4
#define VEC16   (HDIM / 4) // 192 float4 chunks per weight vector
#define NSTAGE  (2 * VEC16)

typedef __attribute__((ext_vector_type(4))) int v4i;
typedef __attribute__((address_space(1))) v4i gv4i;  // global
typedef __attribute__((address_space(3))) v4i lv4i;  // LDS

__device__ __forceinline__ float wave_allreduce_sum(float v) {
#pragma unroll
    for (int off = 16; off > 0; off >>= 1)
        v += __shfl_xor(v, off, 32);
    return v;
}

__global__ __launch_bounds__(256) void token_to_word_agg_kernel(
    const float* __restrict__ feat,    // [T, H]
    const float* __restrict__ attn_w,  // [H]
    const float* __restrict__ attn_b,  // [1]
    const float* __restrict__ wt_w,    // [H]
    const float* __restrict__ wt_b,    // [1]
    const int*   __restrict__ seg,     // [T] sorted
    float*       __restrict__ out,     // [W, H]
    int T, int W)
{
    __shared__ __align__(16) float wsm[2 * HDIM]; // [0,768): attn_w  [768,1536): wt_w

    // ---- block-level async staging of the shared weight vectors into LDS ----
    {
        const int tid = threadIdx.x;
#if __has_builtin(__builtin_amdgcn_global_load_async_to_lds_b128)
        for (int c = tid; c < NSTAGE; c += 256) {
            const float* src = (c < VEC16) ? (attn_w + c * 4) : (wt_w + (c - VEC16) * 4);
            __builtin_amdgcn_global_load_async_to_lds_b128(
                (gv4i*)src, (lv4i*)&wsm[c * 4],
                /*offset=*/0, /*cpol=*/0);
        }
#if __has_builtin(__builtin_amdgcn_s_wait_asynccnt)
        __builtin_amdgcn_s_wait_asynccnt(0);
#else
        asm volatile("s_wait_asynccnt 0x0" ::: "memory");
#endif
#else
        for (int c = tid; c < NSTAGE; c += 256) {
            const float* src = (c < VEC16) ? (attn_w + c * 4) : (wt_w + (c - VEC16) * 4);
            *(float4*)&wsm[c * 4] = *(const float4*)src;
        }
#endif
        __syncthreads();
    }

    const int lane = threadIdx.x & 31;
    const int wid  = (int)((blockIdx.x * blockDim.x + threadIdx.x) >> 5);
    if (wid >= W) return;

    const int colBase = lane * 4;

    // Per-wave weight registers, read once from LDS (ds_load_b128).
    float4 aw[CHUNKS], ww[CHUNKS];
#pragma unroll
    for (int j = 0; j < CHUNKS; ++j) {
        aw[j] = *(const float4*)&wsm[colBase + j * CSTRIDE];
        ww[j] = *(const float4*)&wsm[HDIM + colBase + j * CSTRIDE];
    }
    const float ab = attn_b[0];
    const float wb = wt_b[0];

    // ---- segment start: uniform binary search (lower_bound of wid) ----
    int lo = 0, hi = T;
    while (lo < hi) {
        int mid = (lo + hi) >> 1;
        if (seg[mid] < wid) lo = mid + 1; else hi = mid;
    }
    const int s = lo;

    // Warm the feature stream for this segment (gfx1250 global_prefetch path).
    if (s < T) __builtin_prefetch(feat + (size_t)s * HDIM + colBase, 0, 0);

    // ---- segment end: lane-parallel ballot scan (avg length 2) ----
    int e = s;
    for (;;) {
        int t = e + lane;
        unsigned long long bm = __ballot((t < T) && (seg[t] == wid));
        unsigned v = (unsigned)bm;
        if (v == 0xffffffffu) { e += 32; continue; }
        e += __builtin_ctz(~v);   // contiguous prefix of matches (ids are sorted)
        break;
    }

    float* op = out + (size_t)wid * HDIM + colBase;

    if (s == e) {
        // Empty word: agg = 0 -> agg * gate = 0.
#pragma unroll
        for (int j = 0; j < CHUNKS; ++j) {
            __builtin_nontemporal_store(0.0f, op + j * CSTRIDE + 0);
            __builtin_nontemporal_store(0.0f, op + j * CSTRIDE + 1);
            __builtin_nontemporal_store(0.0f, op + j * CSTRIDE + 2);
            __builtin_nontemporal_store(0.0f, op + j * CSTRIDE + 3);
        }
        return;
    }

    // ---- pass A: exact segment max of scores (matches reference numerics) ----
    float m = -INFINITY;
    for (int t = s; t < e; ++t) {
        const float* fp = feat + (size_t)t * HDIM + colBase;
        float part = 0.0f;
#pragma unroll
        for (int j = 0; j < CHUNKS; ++j) {
            float4 f = *(const float4*)(fp + j * CSTRIDE);
            part = fmaf(f.x, aw[j].x, part);
            part = fmaf(f.y, aw[j].y, part);
            part = fmaf(f.z, aw[j].z, part);
            part = fmaf(f.w, aw[j].w, part);
        }
        float sc = wave_allreduce_sum(part) + ab;
        m = fmaxf(m, sc);
    }

    // ---- pass B: exp-weighted accumulation (segment is L2-resident, ~6 KB) ----
    float d = 0.0f;
    float4 acc[CHUNKS];
#pragma unroll
    for (int j = 0; j < CHUNKS; ++j) acc[j] = make_float4(0.f, 0.f, 0.f, 0.f);

    for (int t = s; t < e; ++t) {
        const float* fp = feat + (size_t)t * HDIM + colBase;
        float4 f[CHUNKS];
        float part = 0.0f;
#pragma unroll
        for (int j = 0; j < CHUNKS; ++j) {
            f[j] = *(const float4*)(fp + j * CSTRIDE);
            part = fmaf(f[j].x, aw[j].x, part);
            part = fmaf(f[j].y, aw[j].y, part);
            part = fmaf(f[j].z, aw[j].z, part);
            part = fmaf(f[j].w, aw[j].w, part);
        }
        float sc = wave_allreduce_sum(part) + ab;
        float w  = __expf(sc - m);
        d += w;
#pragma unroll
        for (int j = 0; j < CHUNKS; ++j) {
            acc[j].x = fmaf(w, f[j].x, acc[j].x);
            acc[j].y = fmaf(w, f[j].y, acc[j].y);
            acc[j].z = fmaf(w, f[j].z, acc[j].z);
            acc[j].w = fmaf(w, f[j].w, acc[j].w);
        }
    }

    const float inv = 1.0f / d;   // non-empty => d >= exp(0) = 1
    float gpart = 0.0f;
#pragma unroll
    for (int j = 0; j < CHUNKS; ++j) {
        acc[j].x *= inv; acc[j].y *= inv; acc[j].z *= inv; acc[j].w *= inv;
        gpart = fmaf(acc[j].x, ww[j].x, gpart);
        gpart = fmaf(acc[j].y, ww[j].y, gpart);
        gpart = fmaf(acc[j].z, ww[j].z, gpart);
        gpart = fmaf(acc[j].w, ww[j].w, gpart);
    }
    const float gl   = wave_allreduce_sum(gpart) + wb;
    const float gate = 1.0f / (1.0f + __expf(-gl));

#pragma unroll
    for (int j = 0; j < CHUNKS; ++j) {
        __builtin_nontemporal_store(acc[j].x * gate, op + j * CSTRIDE + 0);
        __builtin_nontemporal_store(acc[j].y * gate, op + j * CSTRIDE + 1);
        __builtin_nontemporal_store(acc[j].z * gate, op + j * CSTRIDE + 2);
        __builtin_nontemporal_store(acc[j].w * gate, op + j * CSTRIDE + 3);
    }
}

extern "C" void kernel_launch(void* const* d_in, const int* in_sizes, int n_in,
                              void* d_out, int out_size, void* d_ws, size_t ws_size,
                              hipStream_t stream) {
    const float* feat   = (const float*)d_in[0];   // [T, 768]
    const float* attn_w = (const float*)d_in[1];   // [768]
    const float* attn_b = (const float*)d_in[2];   // [1]
    const float* wt_w   = (const float*)d_in[3];   // [768]
    const float* wt_b   = (const float*)d_in[4];   // [1]
    const int*   seg    = (const int*)d_in[5];     // [T]
    float*       out    = (float*)d_out;           // [W, 768]

    const int T = in_sizes[5];
    const int W = out_size / HDIM;

    // one wave32 per word, 8 waves per 256-thread block
    const int blocks = (W + 7) / 8;
    token_to_word_agg_kernel<<<blocks, 256, 0, stream>>>(
        feat, attn_w, attn_b, wt_w, wt_b, seg, out, T, W);
}